// GraphAttentionLayer_77678778515733
// MI455X (gfx1250) — compile-verified
//
#include <hip/hip_runtime.h>
#include <hip/hip_bf16.h>

#define FDIM 512
#define HDIM 256
#define NHEADS 8
#define NEG_SLOPE 0.01f

typedef float v2f __attribute__((ext_vector_type(2)));
typedef float v8f __attribute__((ext_vector_type(8)));
typedef unsigned int u32x4 __attribute__((ext_vector_type(4)));
typedef int i32x4 __attribute__((ext_vector_type(4)));
typedef int i32x8 __attribute__((ext_vector_type(8)));

// ---------------------------------------------------------------------------
// Kernel 1: h = feats @ W_w + W_b  via V_WMMA_F32_16X16X4_F32.
// One block = one 16-row M tile. The 16x512 fp32 A tile (32 KB) is staged
// into LDS once per block by the Tensor Data Mover (tensor_load_to_lds,
// TENSORcnt), then the 8 waves read A-fragments with ds_load_b64 and stream
// B columns from L2-resident W_w. Each wave owns 2 of the 16 column tiles.
// ---------------------------------------------------------------------------
__global__ void __launch_bounds__(256)
k_gemm(const float* __restrict__ A,      // N x 512 feats
       const float* __restrict__ W,      // 512 x 256
       const float* __restrict__ bias,   // 256
       float* __restrict__ Hout,         // N x 256
       int Nn) {
  __shared__ float sA[16 * FDIM];        // 32 KB A tile

  const int mtile = blockIdx.x;
  const int wave  = threadIdx.x >> 5;
  const int lane  = threadIdx.x & 31;
  const int l16   = lane & 15;
  const int hi    = lane >> 4;           // 0: K pair {0,1}, 1: K pair {2,3}

  // ---- TDM stage: wave 0 issues one 2-D tile DMA (16 rows x 512 cols, 4B) --
  if (wave == 0) {
    const unsigned long long gaddr =
        (unsigned long long)(const void*)(A + (size_t)mtile * 16 * FDIM);
    const unsigned lds_off = (unsigned)(unsigned long long)(void*)sA;
    int rows = Nn - mtile * 16;
    if (rows > 16) rows = 16;

    u32x4 g0;
    g0[0] = 1u;                                    // count=1 (valid user D#)
    g0[1] = lds_off;                               // lds_addr (bytes)
    g0[2] = (unsigned)gaddr;                       // global_addr[31:0]
    g0[3] = (unsigned)(gaddr >> 32) | (2u << 30);  // global_addr[56:32]|type=2

    i32x8 g1;
    g1[0] = (2 << 16);         // workgroup_mask=0, data_size=2 (4 bytes)
    g1[1] = (FDIM << 16);      // tensor_dim0[15:0]=512 (bits 63:48)
    g1[2] = (rows << 16);      // tensor_dim0 hi=0 | tensor_dim1[15:0]=rows
    g1[3] = (FDIM << 16);      // tensor_dim1 hi=0 | tile_dim0=512
    g1[4] = rows;              // tile_dim1=rows | tile_dim2=0
    g1[5] = FDIM;              // tensor_dim0_stride=512 (elements)
    g1[6] = 0;                 // stride hi | tensor_dim1_stride lo
    g1[7] = 0;                 // tensor_dim1_stride hi
    i32x4 gz4 = {0, 0, 0, 0};                  // groups 2/3 unused (2-D tile)
    i32x8 gz8 = {0, 0, 0, 0, 0, 0, 0, 0};      // unused second-pair slot

    __builtin_amdgcn_tensor_load_to_lds(g0, g1, gz4, gz4, gz8, 0);
    __builtin_amdgcn_s_wait_tensorcnt(0);
  }
  __syncthreads();

  const int c0 = (wave)     * 16 + l16;  // column for N-tile 0
  const int c1 = (wave + 8) * 16 + l16;  // column for N-tile 1
  const float* sArow = &sA[l16 * FDIM];

  v8f acc0 = {}; v8f acc1 = {};

  for (int k = 0; k < FDIM; k += 4) {
    const int ka = k + hi * 2;
    v2f a = *(const v2f*)(sArow + ka);            // ds_load_b64
    v2f b0; b0.x = W[(size_t)ka * HDIM + c0];
            b0.y = W[(size_t)(ka + 1) * HDIM + c0];
    v2f b1; b1.x = W[(size_t)ka * HDIM + c1];
            b1.y = W[(size_t)(ka + 1) * HDIM + c1];
    acc0 = __builtin_amdgcn_wmma_f32_16x16x4_f32(false, a, false, b0,
                                                 (short)0, acc0, false, false);
    acc1 = __builtin_amdgcn_wmma_f32_16x16x4_f32(false, a, false, b1,
                                                 (short)0, acc1, false, false);
  }

  const float bb0 = bias[c0];
  const float bb1 = bias[c1];
  #pragma unroll
  for (int r = 0; r < 8; ++r) {
    const int orow = mtile * 16 + r + hi * 8;     // D layout: vgpr r, half hi
    if (orow < Nn) {
      Hout[(size_t)orow * HDIM + c0] = acc0[r] + bb0;
      Hout[(size_t)orow * HDIM + c1] = acc1[r] + bb1;
    }
  }
}

// ---------------------------------------------------------------------------
// Kernel 2: per-node attention projections (float4 vectorized)
//   e_s[n,h] = h[n]·a_s[h] + attn_b[h]   (bias folded here)
//   e_d[n,h] = h[n]·a_d[h]
// ---------------------------------------------------------------------------
__global__ void k_nodeproj(const float* __restrict__ Hmat,
                           const float* __restrict__ attw,   // 8 x 512
                           const float* __restrict__ attb,   // 8
                           float* __restrict__ e_s,
                           float* __restrict__ e_d,
                           int Nn) {
  const int t = blockIdx.x * blockDim.x + threadIdx.x;
  if (t >= Nn * NHEADS) return;
  const int n = t >> 3, head = t & 7;
  const float4* hr = (const float4*)(Hmat + (size_t)n * HDIM);
  const float4* as = (const float4*)(attw + head * (2 * HDIM));
  const float4* ad = (const float4*)(attw + head * (2 * HDIM) + HDIM);
  float es = 0.f, ed = 0.f;
  #pragma unroll 4
  for (int i = 0; i < HDIM / 4; ++i) {
    const float4 hv = hr[i];
    const float4 a  = as[i];
    const float4 d  = ad[i];
    es += hv.x * a.x + hv.y * a.y + hv.z * a.z + hv.w * a.w;
    ed += hv.x * d.x + hv.y * d.y + hv.z * d.z + hv.w * d.w;
  }
  e_s[t] = es + attb[head];
  e_d[t] = ed;
}

// ---------------------------------------------------------------------------
// CSR construction: zero -> histogram -> single-block scan -> scatter
// ---------------------------------------------------------------------------
__global__ void k_zero_i32(int* __restrict__ p, int n) {
  const int i = blockIdx.x * blockDim.x + threadIdx.x;
  if (i < n) p[i] = 0;
}

__global__ void k_count(const int* __restrict__ src, int* __restrict__ cnt, int E) {
  const int e = blockIdx.x * blockDim.x + threadIdx.x;
  if (e < E) atomicAdd(&cnt[src[e]], 1);
}

__global__ void __launch_bounds__(1024)
k_scan(const int* __restrict__ cnt, int* __restrict__ row_ptr, int Nn) {
  __shared__ int lds[1024];
  __shared__ int carry;
  const int tid = threadIdx.x;
  if (tid == 0) carry = 0;
  __syncthreads();
  for (int base = 0; base < Nn; base += 1024) {
    const int i = base + tid;
    const int v = (i < Nn) ? cnt[i] : 0;
    lds[tid] = v;
    __syncthreads();
    for (int off = 1; off < 1024; off <<= 1) {
      const int add = (tid >= off) ? lds[tid - off] : 0;
      __syncthreads();
      lds[tid] += add;
      __syncthreads();
    }
    if (i < Nn) row_ptr[i] = carry + lds[tid] - v;   // exclusive
    __syncthreads();
    if (tid == 1023) carry += lds[1023];
    __syncthreads();
  }
  if (tid == 0) row_ptr[Nn] = carry;
}

__global__ void k_scatter(const int* __restrict__ src, const int* __restrict__ dst,
                          const int* __restrict__ row_ptr, int* __restrict__ fill,
                          int* __restrict__ csr_dst, int E) {
  const int e = blockIdx.x * blockDim.x + threadIdx.x;
  if (e >= E) return;
  const int s = src[e];
  const int pos = row_ptr[s] + atomicAdd(&fill[s], 1);
  csr_dst[pos] = dst[e];
}

// ---------------------------------------------------------------------------
// Kernel 3: per-node softmax + aggregation. One block per node, no global
// atomics. h (41 MB) is L2-resident so the h[dst] gathers hit L2.
// ---------------------------------------------------------------------------
__global__ void __launch_bounds__(256)
k_agg(const float* __restrict__ Hmat,
      const float* __restrict__ e_s, const float* __restrict__ e_d,
      const int* __restrict__ row_ptr, const int* __restrict__ csr_dst,
      float* __restrict__ out, int Nn) {
  const int n   = blockIdx.x;
  const int tid = threadIdx.x;
  const int head = tid & 7;
  const int j    = tid >> 3;

  __shared__ float s_es[NHEADS];
  __shared__ float s_m[NHEADS];
  __shared__ float s_den[NHEADS];
  __shared__ float s_alpha[NHEADS];
  __shared__ float s_red[256];

  const int beg = row_ptr[n], end = row_ptr[n + 1];
  if (tid < NHEADS) s_es[tid] = e_s[n * NHEADS + tid];
  __syncthreads();

  // pass 1: per-head max
  float mloc = -__builtin_huge_valf();
  for (int i = beg + j; i < end; i += 32) {
    const int d = csr_dst[i];
    float s = s_es[head] + e_d[d * NHEADS + head];
    s = (s > 0.f) ? s : NEG_SLOPE * s;
    mloc = fmaxf(mloc, s);
  }
  s_red[tid] = mloc;
  __syncthreads();
  for (int off = 128; off >= 8; off >>= 1) {
    if (tid < off) s_red[tid] = fmaxf(s_red[tid], s_red[tid + off]);
    __syncthreads();
  }
  if (tid < NHEADS) s_m[tid] = s_red[tid];
  __syncthreads();

  // pass 2: per-head denom
  const float mh = s_m[head];
  float dloc = 0.f;
  for (int i = beg + j; i < end; i += 32) {
    const int d = csr_dst[i];
    float s = s_es[head] + e_d[d * NHEADS + head];
    s = (s > 0.f) ? s : NEG_SLOPE * s;
    dloc += __expf(s - mh);
  }
  s_red[tid] = dloc;
  __syncthreads();
  for (int off = 128; off >= 8; off >>= 1) {
    if (tid < off) s_red[tid] += s_red[tid + off];
    __syncthreads();
  }
  if (tid < NHEADS) s_den[tid] = s_red[tid];
  __syncthreads();

  // pass 3: accumulate alpha * h[dst]; thread owns column c = tid
  float acc[NHEADS];
  #pragma unroll
  for (int hh = 0; hh < NHEADS; ++hh) acc[hh] = 0.f;

  for (int i = beg; i < end; ++i) {
    const int d = csr_dst[i];
    if (tid < NHEADS) {
      float s = s_es[tid] + e_d[d * NHEADS + tid];
      s = (s > 0.f) ? s : NEG_SLOPE * s;
      s_alpha[tid] = __expf(s - s_m[tid]) / s_den[tid];
    }
    __syncthreads();
    const float hd = Hmat[(size_t)d * HDIM + tid];
    #pragma unroll
    for (int hh = 0; hh < NHEADS; ++hh) acc[hh] += s_alpha[hh] * hd;
    __syncthreads();
  }

  const float hv = Hmat[(size_t)n * HDIM + tid];
  #pragma unroll
  for (int hh = 0; hh < NHEADS; ++hh)
    out[(size_t)n * (NHEADS * HDIM) + hh * HDIM + tid] = hv + acc[hh];
}

// ---------------------------------------------------------------------------
extern "C" void kernel_launch(void* const* d_in, const int* in_sizes, int n_in,
                              void* d_out, int out_size, void* d_ws, size_t ws_size,
                              hipStream_t stream) {
  const float* feats  = (const float*)d_in[0];
  // d_in[1] adjacency_matrix: unused placeholder
  const int*   adj    = (const int*)d_in[2];
  const float* W_w    = (const float*)d_in[3];
  const float* W_b    = (const float*)d_in[4];
  const float* attn_w = (const float*)d_in[5];
  const float* attn_b = (const float*)d_in[6];

  const int Nn = in_sizes[0] / FDIM;
  const int E  = in_sizes[2] / 2;
  const int* srcp = adj;
  const int* dstp = adj + E;

  // workspace layout
  float* Hm      = (float*)d_ws;                      // Nn*256
  float* e_s     = Hm + (size_t)Nn * HDIM;            // Nn*8
  float* e_d     = e_s + (size_t)Nn * NHEADS;         // Nn*8
  int*   cnt     = (int*)(e_d + (size_t)Nn * NHEADS); // Nn
  int*   fill    = cnt + Nn;                          // Nn
  int*   row_ptr = fill + Nn;                         // Nn+1
  int*   csr_dst = row_ptr + Nn + 1;                  // E
  float* outp    = (float*)d_out;

  k_gemm<<<(Nn + 15) / 16, 256, 0, stream>>>(feats, W_w, W_b, Hm, Nn);
  k_nodeproj<<<(Nn * NHEADS + 255) / 256, 256, 0, stream>>>(Hm, attn_w, attn_b,
                                                            e_s, e_d, Nn);
  k_zero_i32<<<(2 * Nn + 255) / 256, 256, 0, stream>>>(cnt, 2 * Nn);
  k_count<<<(E + 255) / 256, 256, 0, stream>>>(srcp, cnt, E);
  k_scan<<<1, 1024, 0, stream>>>(cnt, row_ptr, Nn);
  k_scatter<<<(E + 255) / 256, 256, 0, stream>>>(srcp, dstp, row_ptr, fill,
                                                 csr_dst, E);
  k_agg<<<Nn, 256, 0, stream>>>(Hm, e_s, e_d, row_ptr, csr_dst, outp, Nn);
}